// PositionSelector_76373108457716
// MI455X (gfx1250) — compile-verified
//
#include <hip/hip_runtime.h>

#define BB 64    // batch
#define SS 256   // seq length
#define VV 2048  // vocab
#define HH 512   // hidden

typedef __attribute__((ext_vector_type(2))) float v2f;
typedef __attribute__((ext_vector_type(8))) float v8f;

// -------------------------------------------------------------------------
// Stage 1: h1[b, :] = relu( sum_p W1[p*V + x[b,p], :] + b1 )
// One block per batch row; 128 threads, each owns a float4 (4 columns).
// Token index load is wave-uniform -> scalar load; W1 row load is a fully
// coalesced 2 KB burst per position. This stage is the HBM-bound hot spot.
// -------------------------------------------------------------------------
__global__ __launch_bounds__(128) void gather_sum_relu_kernel(
    const int* __restrict__ x, const float* __restrict__ W1,
    const float* __restrict__ b1, float* __restrict__ h1) {
  const int b   = blockIdx.x;
  const int col = threadIdx.x * 4;

  float4 acc = make_float4(0.f, 0.f, 0.f, 0.f);
  const int* xb = x + b * SS;

  for (int p = 0; p < SS; ++p) {
    const int tok = xb[p];                           // uniform across block
    const long long row = (long long)p * VV + tok;   // row of W1
    const float4 w =
        *reinterpret_cast<const float4*>(W1 + row * (long long)HH + col);
    acc.x += w.x; acc.y += w.y; acc.z += w.z; acc.w += w.w;
  }

  const float4 bias = *reinterpret_cast<const float4*>(b1 + col);
  acc.x = fmaxf(acc.x + bias.x, 0.f);
  acc.y = fmaxf(acc.y + bias.y, 0.f);
  acc.z = fmaxf(acc.z + bias.z, 0.f);
  acc.w = fmaxf(acc.w + bias.w, 0.f);
  *reinterpret_cast<float4*>(h1 + b * HH + col) = acc;
}

// -------------------------------------------------------------------------
// Stages 2 & 3: C[64, N] = (relu?)( A[64, K] @ Bm[K, N] + bias[N] )
// One wave32 computes one 16x16 tile via V_WMMA_F32_16X16X4_F32,
// accumulating K in steps of 4 (K/4 = 128 WMMA issues per tile).
//
// Per CDNA5 ISA VGPR layouts (wave32):
//   A 16x4 f32 : lanes 0-15 -> M=lane, {K=0,K=1} in v0,v1
//                lanes 16-31 -> M=lane-16, {K=2,K=3}
//   B 4x16 f32 : lanes 0-15 -> N=lane, {K=0,K=1}
//                lanes 16-31 -> N=lane-16, {K=2,K=3}
//   C/D 16x16  : vgpr r, lanes 0-15 -> (M=r,    N=lane)
//                vgpr r, lanes 16-31 -> (M=r+8, N=lane-16)
// -------------------------------------------------------------------------
template <bool RELU>
__global__ __launch_bounds__(128) void wmma_gemm_kernel(
    const float* __restrict__ A, const float* __restrict__ Bm,
    const float* __restrict__ bias, float* __restrict__ C,
    int N, int K) {
  const int tilesN = N >> 4;
  const int gid    = blockIdx.x * 4 + (threadIdx.x >> 5);  // global wave id
  const int tileM  = (gid / tilesN) << 4;
  const int tileN  = (gid % tilesN) << 4;

  const int lane = threadIdx.x & 31;
  const int half = lane >> 4;   // 0: K pair {0,1}; 1: K pair {2,3}
  const int l    = lane & 15;   // M index for A, N index for B/C/D

  v8f acc = {};

  const float* arow = A + (tileM + l) * K;   // row of A owned by this lane

  for (int k = 0; k < K; k += 4) {
    const int ka = k + half * 2;

    // A fragment: two consecutive K values -> aligned float2 load
    const float2 av = *reinterpret_cast<const float2*>(arow + ka);
    v2f a; a.x = av.x; a.y = av.y;

    // B fragment: rows ka, ka+1 of Bm at column tileN + l
    v2f bf;
    bf.x = Bm[(long long)(ka    ) * N + tileN + l];
    bf.y = Bm[(long long)(ka + 1) * N + tileN + l];

    acc = __builtin_amdgcn_wmma_f32_16x16x4_f32(
        /*neg_a=*/false, a, /*neg_b=*/false, bf,
        /*c_mod=*/(short)0, acc, /*reuse_a=*/false, /*reuse_b=*/false);
  }

  const int col  = tileN + l;
  const int row0 = tileM + half * 8;
  const float bv = bias[col];

#pragma unroll
  for (int r = 0; r < 8; ++r) {
    float v = acc[r] + bv;
    if (RELU) v = fmaxf(v, 0.f);
    C[(row0 + r) * N + col] = v;
  }
}

// -------------------------------------------------------------------------
extern "C" void kernel_launch(void* const* d_in, const int* in_sizes, int n_in,
                              void* d_out, int out_size, void* d_ws, size_t ws_size,
                              hipStream_t stream) {
  const int*   x  = (const int*)d_in[0];
  const float* W1 = (const float*)d_in[1];
  const float* b1 = (const float*)d_in[2];
  const float* W2 = (const float*)d_in[3];
  const float* b2 = (const float*)d_in[4];
  const float* W3 = (const float*)d_in[5];
  const float* b3 = (const float*)d_in[6];
  float* out = (float*)d_out;

  float* h1 = (float*)d_ws;          // [64, 512]
  float* h2 = h1 + BB * HH;          // [64, 512]

  // Stage 1: gather-sum + bias + relu -> h1
  gather_sum_relu_kernel<<<BB, 128, 0, stream>>>(x, W1, b1, h1);

  // Stage 2: h2 = relu(h1 @ W2 + b2), M=64 N=512 K=512
  // tiles = (64/16)*(512/16) = 128 waves -> 32 blocks of 4 waves
  wmma_gemm_kernel<true><<<(BB / 16) * (HH / 16) / 4, 128, 0, stream>>>(
      h1, W2, b2, h2, HH, HH);

  // Stage 3: out = h2 @ W3 + b3, M=64 N=256 K=512
  // tiles = (64/16)*(256/16) = 64 waves -> 16 blocks of 4 waves
  wmma_gemm_kernel<false><<<(BB / 16) * (SS / 16) / 4, 128, 0, stream>>>(
      h2, W3, b3, out, SS, HH);
}